// MultiheadAttention_Residual_88192858456085
// MI455X (gfx1250) — compile-verified
//
#include <hip/hip_runtime.h>

typedef float v2f __attribute__((ext_vector_type(2)));
typedef float v8f __attribute__((ext_vector_type(8)));

#define B_  4
#define L_  2048
#define D_  512
#define M_  1200
#define H_  6
#define D4_ 2048
#define EPS_ 1e-5f

// workspace layout (float offsets)
#define WS_VV 0        // B*M   = 4800
#define WS_OV 4800     // B*D   = 2048
#define WS_WC 8192     // D*D   = 262144
#define WS_BC 270336   // D     = 512

// ---------------------------------------------------------------------------
// sm output is exactly uniform 1/L (softmax over constant logits): pure fill.
// ---------------------------------------------------------------------------
__global__ void k_fill_sm(float4* __restrict__ p, size_t n4) {
    const float v = 1.0f / 2048.0f;      // exact power of two
    const float4 q = make_float4(v, v, v, v);
    size_t stride = (size_t)gridDim.x * blockDim.x;
    for (size_t i = (size_t)blockIdx.x * blockDim.x + threadIdx.x; i < n4; i += stride)
        p[i] = q;
}

// V_vec[b,m] = sum_d keyval[b,d] * Wkv[d, M+m] + bkv[M+m]
__global__ void k_vvec(const float* __restrict__ keyval, const float* __restrict__ Wkv,
                       const float* __restrict__ bkv, float* __restrict__ Vv) {
    int t = blockIdx.x * 256 + threadIdx.x;
    if (t >= B_ * M_) return;
    int b = t / M_, m = t - b * M_;
    const float* x = keyval + (size_t)b * D_;
    const float* w = Wkv + (M_ + m);
    float s = bkv[M_ + m];
    for (int d = 0; d < D_; ++d) s += x[d] * w[(size_t)d * (2 * M_)];
    Vv[t] = s;
}

// out_vec[b,n] = sum_m V_vec[b,m] * Wff[m,n] + bff[n]
__global__ void k_ovec(const float* __restrict__ Vv, const float* __restrict__ Wff,
                       const float* __restrict__ bff, float* __restrict__ Ov) {
    int t = blockIdx.x * 256 + threadIdx.x;
    if (t >= B_ * D_) return;
    int b = t >> 9, n = t & (D_ - 1);
    float s = bff[n];
    const float* v = Vv + (size_t)b * M_;
    for (int m = 0; m < M_; ++m) s += v[m] * Wff[(size_t)m * D_ + n];
    Ov[t] = s;
}

// bc[n] = sum_k b2a[k] * W2b[k,n] + b2b[n]   (leaky slope 1.0 == identity)
__global__ void k_bc(const float* __restrict__ b2a, const float* __restrict__ W2b,
                     const float* __restrict__ b2b, float* __restrict__ bc) {
    int t = blockIdx.x * 256 + threadIdx.x;
    if (t >= D_) return;
    float s = b2b[t];
    for (int k = 0; k < D4_; ++k) s += b2a[k] * W2b[(size_t)k * D_ + t];
    bc[t] = s;
}

// ---------------------------------------------------------------------------
// Wc = W2a (512x2048) @ W2b (2048x512), fp32 WMMA, one wave per 16x16 C tile.
// ---------------------------------------------------------------------------
__global__ void __launch_bounds__(256) k_wc(const float* __restrict__ W2a,
                                            const float* __restrict__ W2b,
                                            float* __restrict__ Wc) {
    int tid  = threadIdx.x;
    int lane = tid & 31, w = tid >> 5;
    int tile = blockIdx.x * 8 + w;           // 0..1023 (32x32 tiles)
    int tn = tile & 31, tm = tile >> 5;
    int mb = tm * 16, nb = tn * 16;
    int half = lane >> 4, lm = lane & 15;

    v8f acc = {};
    const float* Arow = W2a + (size_t)(mb + lm) * D4_;
    for (int k0 = 0; k0 < D4_; k0 += 4) {
        v2f a, bf;
        a.x  = Arow[k0 + 2 * half];
        a.y  = Arow[k0 + 2 * half + 1];
        const float* Brow = W2b + (size_t)(k0 + 2 * half) * D_ + nb + lm;
        bf.x = Brow[0];
        bf.y = Brow[D_];
        acc = __builtin_amdgcn_wmma_f32_16x16x4_f32(false, a, false, bf,
                                                    (short)0, acc, false, false);
    }
#pragma unroll
    for (int vr = 0; vr < 8; ++vr)
        Wc[(size_t)(mb + vr + 8 * half) * D_ + nb + lm] = acc[vr];
}

// ---------------------------------------------------------------------------
// Fused per-token kernel: LN1(query + out_vec) -> GEMM(x @ Wc + bc) via WMMA
// -> residual -> LN2 -> + keyval broadcast -> out3.
// One workgroup (8 waves) per 16 tokens; each wave owns a 64-column slice.
// B tiles are double-buffered through LDS with CDNA5 async global->LDS loads
// (per-wave ownership => only s_wait_asynccnt needed, no barriers in MAC loop).
// ---------------------------------------------------------------------------
__global__ void __launch_bounds__(256) k_main(const float* __restrict__ query,
                                              const float* __restrict__ keyval,
                                              const float* __restrict__ g1,
                                              const float* __restrict__ b1,
                                              const float* __restrict__ g2,
                                              const float* __restrict__ b2,
                                              const float* __restrict__ Ov,
                                              const float* __restrict__ Wc,
                                              const float* __restrict__ bc,
                                              float* __restrict__ out3) {
    __shared__ float sA[16][516];            // 516 = 512 + pad (bank-conflict dodge)
    __shared__ float sB[8][2][256];          // per-wave double-buffered 4x64 B tile
    __shared__ float red1[256], red2[256];
    __shared__ float rowm[16], rowi[16];

    int tid = threadIdx.x;
    int r = tid >> 4, j = tid & 15;          // row / 16-thread column group
    int blk = blockIdx.x;
    int b = blk >> 7;                        // 128 blocks per batch
    int row0 = blk * 16;

    // ---- Phase 1: x = query + out_vec[b]; LN1 into LDS ----
    const float* q  = query + (size_t)(row0 + r) * D_;
    const float* ov = Ov + (size_t)b * D_;
    float s = 0.f, s2 = 0.f;
    for (int i = 0; i < 32; ++i) {
        int c = j + 16 * i;
        float x = q[c] + ov[c];
        sA[r][c] = x;
        s += x; s2 += x * x;
    }
    red1[tid] = s; red2[tid] = s2;
    __syncthreads();
    if (tid < 16) {
        float su = 0.f, sq = 0.f;
        for (int k = 0; k < 16; ++k) { su += red1[tid * 16 + k]; sq += red2[tid * 16 + k]; }
        float mean = su * (1.0f / D_);
        float var  = sq * (1.0f / D_) - mean * mean;
        rowm[tid] = mean;
        rowi[tid] = rsqrtf(var + EPS_);
    }
    __syncthreads();
    for (int i = 0; i < 32; ++i) {
        int c = j + 16 * i;
        sA[r][c] = (sA[r][c] - rowm[r]) * rowi[r] * g1[c] + b1[c];
    }
    __syncthreads();

    // ---- Phase 2: y = out_ @ Wc via fp32 WMMA; B streamed with async->LDS ----
    int lane = tid & 31, w = tid >> 5;
    int half = lane >> 4, lm = lane & 15;

    // async-load geometry: 2 x b128 per lane covers a 4x64 chunk for this wave
    const int r0 = lane >> 4;                // k-row within pair: 0/1
    const int c0 = (lane & 15) * 4;          // 4 consecutive columns per lane
    const float* gbase = Wc + (size_t)r0 * D_ + (size_t)w * 64 + c0;
    uint32_t lds0 = (uint32_t)(uintptr_t)(&sB[w][0][0]) + (uint32_t)(r0 * 64 + c0) * 4u;
    const uint32_t BUFSTRIDE = 256u * 4u;    // bytes between double buffers
    const uint32_t ROWPAIR   = 2u * 64u * 4u;

#define ISSUE_CHUNK(K0, BUF)                                                     \
    do {                                                                         \
        const float* _g0 = gbase + (size_t)(K0) * D_;                            \
        const float* _g1 = _g0 + 2 * D_;                                         \
        uint32_t _l0 = lds0 + (uint32_t)(BUF) * BUFSTRIDE;                       \
        uint32_t _l1 = _l0 + ROWPAIR;                                            \
        asm volatile("global_load_async_to_lds_b128 %0, %1, off"                 \
                     :: "v"(_l0), "v"(_g0) : "memory");                          \
        asm volatile("global_load_async_to_lds_b128 %0, %1, off"                 \
                     :: "v"(_l1), "v"(_g1) : "memory");                          \
    } while (0)

    v8f acc[4] = {};
    ISSUE_CHUNK(0, 0);
    int buf = 0;
    for (int k0 = 0; k0 < D_; k0 += 4) {
        if (k0 + 4 < D_) {
            ISSUE_CHUNK(k0 + 4, buf ^ 1);
            // oldest chunk (2 loads) complete; next chunk (2 loads) in flight
            asm volatile("s_wait_asynccnt 0x2" ::: "memory");
        } else {
            asm volatile("s_wait_asynccnt 0x0" ::: "memory");
        }
        v2f a;
        a.x = sA[lm][k0 + 2 * half];
        a.y = sA[lm][k0 + 2 * half + 1];
        const float* Bb = &sB[w][buf][0];
#pragma unroll
        for (int t = 0; t < 4; ++t) {
            v2f bf;
            bf.x = Bb[(2 * half) * 64 + t * 16 + lm];
            bf.y = Bb[(2 * half + 1) * 64 + t * 16 + lm];
            acc[t] = __builtin_amdgcn_wmma_f32_16x16x4_f32(false, a, false, bf,
                                                           (short)0, acc[t], false, false);
        }
        buf ^= 1;
    }
#undef ISSUE_CHUNK
    __syncthreads();

    // ---- Phase 3: t = out_ + y + bc (in-place in LDS) ----
#pragma unroll
    for (int t = 0; t < 4; ++t) {
        int n = w * 64 + t * 16 + lm;
        float bcv = bc[n];
#pragma unroll
        for (int vr = 0; vr < 8; ++vr) {
            int rr = vr + 8 * half;
            sA[rr][n] += acc[t][vr] + bcv;
        }
    }
    __syncthreads();

    // ---- Phase 4: LN2, add broadcast keyval, store out3 ----
    s = 0.f; s2 = 0.f;
    for (int i = 0; i < 32; ++i) {
        int c = j + 16 * i;
        float v = sA[r][c];
        s += v; s2 += v * v;
    }
    red1[tid] = s; red2[tid] = s2;
    __syncthreads();
    if (tid < 16) {
        float su = 0.f, sq = 0.f;
        for (int k = 0; k < 16; ++k) { su += red1[tid * 16 + k]; sq += red2[tid * 16 + k]; }
        float mean = su * (1.0f / D_);
        float var  = sq * (1.0f / D_) - mean * mean;
        rowm[tid] = mean;
        rowi[tid] = rsqrtf(var + EPS_);
    }
    __syncthreads();
    const float* kvb = keyval + (size_t)b * D_;
    float* o = out3 + (size_t)(row0 + r) * D_;
    for (int i = 0; i < 32; ++i) {
        int c = j + 16 * i;
        o[c] = (sA[r][c] - rowm[r]) * rowi[r] * g2[c] + b2[c] + kvb[c];
    }
}

// ---------------------------------------------------------------------------
extern "C" void kernel_launch(void* const* d_in, const int* in_sizes, int n_in,
                              void* d_out, int out_size, void* d_ws, size_t ws_size,
                              hipStream_t stream) {
    (void)in_sizes; (void)n_in; (void)out_size; (void)ws_size;
    const float* query  = (const float*)d_in[0];
    const float* keyval = (const float*)d_in[1];
    // d_in[2]=Wq, d_in[3]=bq : dead (softmax is uniform regardless of logits)
    const float* Wkv = (const float*)d_in[4];
    const float* bkv = (const float*)d_in[5];
    const float* Wff = (const float*)d_in[6];
    const float* bff = (const float*)d_in[7];
    const float* g1  = (const float*)d_in[8];
    const float* b1  = (const float*)d_in[9];
    const float* W2a = (const float*)d_in[10];
    const float* b2a = (const float*)d_in[11];
    const float* W2b = (const float*)d_in[12];
    const float* b2b = (const float*)d_in[13];
    const float* g2  = (const float*)d_in[14];
    const float* b2  = (const float*)d_in[15];

    float* out = (float*)d_out;
    float* ws  = (float*)d_ws;
    float* Vv = ws + WS_VV;
    float* Ov = ws + WS_OV;
    float* Wc = ws + WS_WC;
    float* bc = ws + WS_BC;

    const size_t OUT3 = (size_t)B_ * L_ * D_;            // 4,194,304
    const size_t SM   = (size_t)B_ * H_ * L_ * L_;       // 100,663,296

    // sm output: exact uniform 1/2048 fill (bandwidth-bound, 402 MB)
    k_fill_sm<<<8192, 256, 0, stream>>>((float4*)(out + OUT3), SM / 4);

    // tiny per-batch precomputes
    k_vvec<<<(B_ * M_ + 255) / 256, 256, 0, stream>>>(keyval, Wkv, bkv, Vv);
    k_ovec<<<(B_ * D_) / 256, 256, 0, stream>>>(Vv, Wff, bff, Ov);
    k_bc<<<2, 256, 0, stream>>>(b2a, W2b, b2b, bc);

    // collapse identity-leaky FFN: Wc = W2a @ W2b (fp32 WMMA)
    k_wc<<<128, 256, 0, stream>>>(W2a, W2b, Wc);

    // fused LN1 -> async-B-staged WMMA GEMM -> residual -> LN2 -> +kv
    k_main<<<512, 256, 0, stream>>>(query, keyval, g1, b1, g2, b2, Ov, Wc, bc, out);
}